// EwaldPotential_60601988546635
// MI455X (gfx1250) — compile-verified
//
#include <hip/hip_runtime.h>
#include <math.h>

// ---------------- static problem configuration (matches reference) ----------
#define BFR   4            // frames
#define NATOM 4000         // atoms per frame
#define BN    (BFR*NATOM)  // 16000
#define AK    29           // kx index 0..28
#define CK    57           // ky/kz index -28..28 -> 57
#define MROWS (AK*CK)      // 1653 rows of the GEMM (a,b) pairs
#define KC    40           // K-chunk staged in LDS (multiple of 4, divides 4000)

#define TWOPI_F    6.28318530717958647692f
#define TWOPI_SQ_F 39.4784176043574344753f
#define KSQ_MAX_F  9.86960440108935861883f   // (2*pi/dl)^2, dl=2
#define SELF_COEF  0.06349363593424097f      // 1/(sigma*(2pi)^1.5), sigma=1

typedef __attribute__((ext_vector_type(2))) float v2f;
typedef __attribute__((ext_vector_type(8))) float v8f;

// ---------------------------------------------------------------------------
// Stage 0: zero the 4 output floats (harness poisons d_out with 0xAA).
// ---------------------------------------------------------------------------
__global__ void ewald_zero_out(float* out) {
    if (threadIdx.x < BFR) out[threadIdx.x] = 0.0f;
}

// ---------------------------------------------------------------------------
// Stage 1: per-atom plane-wave tables via complex recurrence.
//   Xr/Xi : [BN][AK]   e^{i 2pi a rx},  a = 0..28
//   Yr/Yi : [BN][CK]   e^{i 2pi b ry},  b index 0..56 meaning b-28
//   Zr/Zi : [BN][CK]   q_n * e^{i 2pi c rz}   (charge folded into Z)
// Also accumulates the Gaussian self-interaction term into out[f].
// ---------------------------------------------------------------------------
__global__ void ewald_tables(const float* __restrict__ pos,
                             const float* __restrict__ q,
                             const float* __restrict__ cell,
                             float* __restrict__ Xr, float* __restrict__ Xi,
                             float* __restrict__ Yr, float* __restrict__ Yi,
                             float* __restrict__ Zr, float* __restrict__ Zi,
                             float* __restrict__ out) {
    int idx = blockIdx.x * blockDim.x + threadIdx.x;
    if (idx >= BN) return;
    int f = idx / NATOM;
    float bx = cell[f*9 + 0], by = cell[f*9 + 4], bz = cell[f*9 + 8];
    float rx = pos[3*idx + 0] / bx;
    float ry = pos[3*idx + 1] / by;
    float rz = pos[3*idx + 2] / bz;
    float qv = q[idx];

    float sx, cx, sy, cy, sz, cz;
    __sincosf(TWOPI_F * rx, &sx, &cx);
    __sincosf(TWOPI_F * ry, &sy, &cy);
    __sincosf(TWOPI_F * rz, &sz, &cz);

    // X: a = 0..28 (hemisphere, kx >= 0)
    {
        size_t xb = (size_t)idx * AK;
        float cr = 1.0f, ci = 0.0f;
        #pragma unroll 1
        for (int a = 0; a < AK; ++a) {
            Xr[xb + a] = cr; Xi[xb + a] = ci;
            float nr = cr*cx - ci*sx;
            ci = cr*sx + ci*cx; cr = nr;
        }
    }
    // Y: b = -28..28 ; negative frequencies are conjugates
    {
        size_t yb = (size_t)idx * CK;
        float cr = 1.0f, ci = 0.0f;
        #pragma unroll 1
        for (int k = 0; k < AK; ++k) {
            Yr[yb + 28 + k] = cr;  Yi[yb + 28 + k] = ci;
            Yr[yb + 28 - k] = cr;  Yi[yb + 28 - k] = -ci;
            float nr = cr*cy - ci*sy;
            ci = cr*sy + ci*cy; cr = nr;
        }
    }
    // Z: c = -28..28, scaled by q_n  (S = sum_n (X*Y)*(q*Z) is unchanged)
    {
        size_t zb = (size_t)idx * CK;
        float cr = 1.0f, ci = 0.0f;
        #pragma unroll 1
        for (int k = 0; k < AK; ++k) {
            Zr[zb + 28 + k] = qv*cr;  Zi[zb + 28 + k] = qv*ci;
            Zr[zb + 28 - k] = qv*cr;  Zi[zb + 28 - k] = -qv*ci;
            float nr = cr*cz - ci*sz;
            ci = cr*sz + ci*cz; cr = nr;
        }
    }

    atomicAdd(&out[f], -qv * qv * SELF_COEF);
}

// ---------------------------------------------------------------------------
// Stage 2: complex GEMM  S[(a,b), c] = sum_n (X[n,a] Y[n,b]) * (q_n Z[n,c])
// on the tensor pipe via V_WMMA_F32_16X16X4_F32, fused with the
// kfac * factor * |S|^2 reduction into out[f].
//
// Four independent accumulators avoid per-iteration sign flips:
//   P1 = sum Are*Bre, P2 = sum Aim*Bim, P3 = sum Are*Bim, P4 = sum Aim*Bre
//   S.re = P1 - P2, S.im = P3 + P4    (combined once in the epilogue)
//
// Fragment layouts (ISA 7.12.2, wave32):
//   A 16x4 : lane L -> row M = L&15, VGPR j -> K = 2*(L>>4)+j
//   B 4x16 : lane L -> col N = L&15, VGPR j -> K = 2*(L>>4)+j
//   C/D    : lane L -> col N = L&15, VGPR v -> row M = v + 8*(L>>4)
// ---------------------------------------------------------------------------
__global__ void __launch_bounds__(256)
ewald_sk_wmma(const float* __restrict__ cell,
              const float* __restrict__ Xr, const float* __restrict__ Xi,
              const float* __restrict__ Yr, const float* __restrict__ Yi,
              const float* __restrict__ Zr, const float* __restrict__ Zi,
              float* __restrict__ out) {
    __shared__ float sZr[KC * 16];
    __shared__ float sZi[KC * 16];

    const int tid  = threadIdx.x;
    const int lane = tid & 31;
    const int wave = tid >> 5;
    const int hi   = lane >> 4;   // 0 or 1
    const int lo   = lane & 15;

    const int f     = blockIdx.z;
    const int n0    = blockIdx.y * 16;          // column tile base (c index)
    const int mtile = blockIdx.x * 8 + wave;    // 0..103
    const int m0    = mtile * 16;

    const int row  = m0 + lo;
    const int rowc = (row < MROWS) ? row : (MROWS - 1);  // clamp; masked later
    const int a    = rowc / CK;
    const int b    = rowc % CK;

    const size_t fN = (size_t)f * NATOM;
    // per-lane column bases into the planar tables; step per-n is AK / CK
    const float* xr0 = Xr + fN * AK + a;
    const float* xi0 = Xi + fN * AK + a;
    const float* yr0 = Yr + fN * CK + b;
    const float* yi0 = Yi + fN * CK + b;

    v8f p1 = {}, p2 = {}, p3 = {}, p4 = {};

    for (int k0 = 0; k0 < NATOM; k0 += KC) {
        __syncthreads();
        // stage (q*Z) tile [KC x 16] into LDS (shared by all 8 waves)
        for (int t = tid; t < KC * 16; t += 256) {
            int kk = t >> 4, cc = t & 15;
            int c = n0 + cc;
            float vr = 0.0f, vi = 0.0f;
            if (c < CK) {
                size_t zi = (fN + (size_t)(k0 + kk)) * CK + c;
                vr = Zr[zi]; vi = Zi[zi];
            }
            sZr[t] = vr; sZi[t] = vi;
        }
        __syncthreads();

        #pragma unroll 4
        for (int kk = 0; kk < KC; kk += 4) {
            v2f Are, Aim, Bre, Bim;
            #pragma unroll
            for (int j = 0; j < 2; ++j) {
                int nl = kk + 2*hi + j;                 // local k in [kk, kk+3]
                size_t n = (size_t)(k0 + nl);
                float xr = xr0[n * AK], xi = xi0[n * AK];
                float yr = yr0[n * CK], yi = yi0[n * CK];
                Are[j] = xr*yr - xi*yi;                 // X*Y complex
                Aim[j] = xr*yi + xi*yr;
                Bre[j] = sZr[nl*16 + lo];
                Bim[j] = sZi[nl*16 + lo];
            }
            p1 = __builtin_amdgcn_wmma_f32_16x16x4_f32(
                false, Are, false, Bre, (short)0, p1, false, false);
            p2 = __builtin_amdgcn_wmma_f32_16x16x4_f32(
                false, Aim, false, Bim, (short)0, p2, false, false);
            p3 = __builtin_amdgcn_wmma_f32_16x16x4_f32(
                false, Are, false, Bim, (short)0, p3, false, false);
            p4 = __builtin_amdgcn_wmma_f32_16x16x4_f32(
                false, Aim, false, Bre, (short)0, p4, false, false);
        }
    }

    // Fused epilogue: sum kfac(k) * factor(a) * |S|^2 over this tile.
    float bx = cell[f*9 + 0], by = cell[f*9 + 4], bz = cell[f*9 + 8];
    float inv_bx = 1.0f / bx, inv_by = 1.0f / by, inv_bz = 1.0f / bz;
    float inv_vol = inv_bx * inv_by * inv_bz;
    float sum = 0.0f;
    #pragma unroll
    for (int v = 0; v < 8; ++v) {
        int m = m0 + v + 8*hi;
        int c = n0 + lo;
        if (m < MROWS && c < CK) {
            int aa = m / CK;
            int bb = (m % CK) - 28;
            int cc = c - 28;
            float gx = (float)aa * inv_bx;
            float gy = (float)bb * inv_by;
            float gz = (float)cc * inv_bz;
            float ksq = TWOPI_SQ_F * (gx*gx + gy*gy + gz*gz);
            if (ksq > 0.0f && ksq <= KSQ_MAX_F) {
                float kfac = __expf(-0.5f * ksq) / ksq;   // sigma^2/2 = 0.5
                float fac  = (aa == 0) ? 1.0f : 2.0f;     // hemisphere weight
                float re = p1[v] - p2[v];
                float im = p3[v] + p4[v];
                sum += fac * kfac * (re*re + im*im);
            }
        }
    }
    sum *= inv_vol;
    // wave32 reduction
    for (int off = 16; off > 0; off >>= 1)
        sum += __shfl_down(sum, off, 32);
    if (lane == 0) atomicAdd(&out[f], sum);
}

// ---------------------------------------------------------------------------
// Workspace layout (floats):  Xr Xi  (BN*AK each)  Yr Yi Zr Zi (BN*CK each)
// Total = BN*(2*29 + 4*57)*4 B ~= 18.3 MB  (fits L2; ws is harness scratch)
// ---------------------------------------------------------------------------
extern "C" void kernel_launch(void* const* d_in, const int* in_sizes, int n_in,
                              void* d_out, int out_size, void* d_ws, size_t ws_size,
                              hipStream_t stream) {
    (void)in_sizes; (void)n_in; (void)out_size; (void)ws_size;
    const float* pos  = (const float*)d_in[0];
    const float* q    = (const float*)d_in[1];
    const float* cell = (const float*)d_in[2];
    // d_in[3] (batch) unused: block-sorted with equal counts by construction.
    float* out = (float*)d_out;

    float* ws = (float*)d_ws;
    size_t XA = (size_t)BN * AK;
    size_t YC = (size_t)BN * CK;
    float* Xr = ws;
    float* Xi = Xr + XA;
    float* Yr = Xi + XA;
    float* Yi = Yr + YC;
    float* Zr = Yi + YC;
    float* Zi = Zr + YC;

    ewald_zero_out<<<1, 32, 0, stream>>>(out);
    ewald_tables<<<(BN + 255) / 256, 256, 0, stream>>>(
        pos, q, cell, Xr, Xi, Yr, Yi, Zr, Zi, out);
    dim3 grid(13 /* 104 m-tiles / 8 waves */, 4 /* ceil(57/16) */, BFR);
    ewald_sk_wmma<<<grid, 256, 0, stream>>>(
        cell, Xr, Xi, Yr, Yi, Zr, Zi, out);
}